// PointNetSharpSeg_19473381720495
// MI455X (gfx1250) — compile-verified
//
#include <hip/hip_runtime.h>
#include <hip/hip_bf16.h>

// ---------------------------------------------------------------------------
// PointNetSharpSeg for MI455X (gfx1250, wave32, WMMA)
// B=8 clouds, N=4096 pts, K=20 NN, channels 4 -> 64 -> 128 -> 256 -> 256
// seg head 256 -> 256 -> 128 -> 5.
// All heavy GEMMs use V_WMMA_F32_16X16X4_F32 (exact fp32 accumulate).
// EdgeConv A-tiles are double-buffered in LDS so gather latency for slot t+1
// hides behind the WMMA K-loop for slot t.
// ---------------------------------------------------------------------------

#define NUM_B   8
#define NUM_N   4096
#define BNPTS   (NUM_B * NUM_N)   // 32768
#define KNN     20
#define EPSV    1e-5f
#define MAXE    17                // ceil(max Kd4 (260) / 16)

typedef float v2f __attribute__((ext_vector_type(2)));
typedef float v8f __attribute__((ext_vector_type(8)));

__device__ __forceinline__ v8f wmma_f32_16x16x4(v2f a, v2f b, v8f c) {
  // 8 args: (neg_a, A, neg_b, B, c_mod, C, reuse_a, reuse_b)
  return __builtin_amdgcn_wmma_f32_16x16x4_f32(false, a, false, b, (short)0, c,
                                               false, false);
}

// ---------------------------------------------------------------------------
// 1) kNN: one thread per query point; register-resident sorted top-20;
//    candidate positions tiled through LDS. grid = B * (N/256), block = 256.
// ---------------------------------------------------------------------------
__global__ __launch_bounds__(256) void knn_kernel(const float* __restrict__ pos,
                                                  int* __restrict__ nbr) {
  __shared__ float sx[256], sy[256], sz[256];
  const int bpc = NUM_N / 256;                                  // 16
  const int b   = blockIdx.x / bpc;
  const int q   = b * NUM_N + (blockIdx.x % bpc) * 256 + threadIdx.x;
  const float qx = pos[q * 3 + 0], qy = pos[q * 3 + 1], qz = pos[q * 3 + 2];

  float kd[KNN]; int ki[KNN];
#pragma unroll
  for (int i = 0; i < KNN; ++i) { kd[i] = 3.0e38f; ki[i] = 0; }

  for (int tile = 0; tile < NUM_N; tile += 256) {
    const int cg = b * NUM_N + tile + threadIdx.x;
    sx[threadIdx.x] = pos[cg * 3 + 0];
    sy[threadIdx.x] = pos[cg * 3 + 1];
    sz[threadIdx.x] = pos[cg * 3 + 2];
    __syncthreads();
    for (int j = 0; j < 256; ++j) {
      const int cand = b * NUM_N + tile + j;
      const float dx = sx[j] - qx, dy = sy[j] - qy, dz = sz[j] - qz;
      const float d = dx * dx + dy * dy + dz * dz;
      if (cand != q && d < kd[KNN - 1]) {
        float okd[KNN]; int oki[KNN];
#pragma unroll
        for (int p = 0; p < KNN; ++p) { okd[p] = kd[p]; oki[p] = ki[p]; }
#pragma unroll
        for (int p = KNN - 1; p >= 1; --p) {
          if (d < okd[p - 1])    { kd[p] = okd[p - 1]; ki[p] = oki[p - 1]; }
          else if (d < okd[p])   { kd[p] = d;          ki[p] = cand;       }
        }
        if (d < okd[0]) { kd[0] = d; ki[0] = cand; }
      }
    }
    __syncthreads();
  }
#pragma unroll
  for (int i = 0; i < KNN; ++i) nbr[q * KNN + i] = ki[i];
}

// ---------------------------------------------------------------------------
// 2) BatchNorm statistics: one WG per channel; single pass sum/sumsq;
//    emits fused scale = g*rsqrt(var+eps), shift = b - mean*scale.
// ---------------------------------------------------------------------------
__global__ __launch_bounds__(256) void bn_stats_kernel(
    const float* __restrict__ h, const float* __restrict__ g,
    const float* __restrict__ beta, float* __restrict__ scale,
    float* __restrict__ shift, int C) {
  __shared__ float rs[256], rq[256];
  const int c = blockIdx.x;
  float s = 0.f, s2 = 0.f;
  for (int r = threadIdx.x; r < BNPTS; r += 256) {
    const float v = h[(size_t)r * C + c];
    s += v; s2 += v * v;
  }
  rs[threadIdx.x] = s; rq[threadIdx.x] = s2;
  __syncthreads();
  for (int off = 128; off > 0; off >>= 1) {
    if (threadIdx.x < off) {
      rs[threadIdx.x] += rs[threadIdx.x + off];
      rq[threadIdx.x] += rq[threadIdx.x + off];
    }
    __syncthreads();
  }
  if (threadIdx.x == 0) {
    const float m   = rs[0] * (1.0f / BNPTS);
    const float var = rq[0] * (1.0f / BNPTS) - m * m;
    const float sc  = g[c] * rsqrtf(var + EPSV);
    scale[c] = sc;
    shift[c] = beta[c] - m * sc;
  }
}

// ---------------------------------------------------------------------------
// 3) Fused BN apply + ReLU (elementwise).
// ---------------------------------------------------------------------------
__global__ __launch_bounds__(256) void bn_relu_kernel(
    const float* __restrict__ h, const float* __restrict__ scale,
    const float* __restrict__ shift, float* __restrict__ o, int C) {
  const int i = blockIdx.x * 256 + threadIdx.x;
  if (i < BNPTS * C) {
    const int c = i % C;
    const float v = fmaf(h[i], scale[c], shift[c]);
    o[i] = v > 0.f ? v : 0.f;
  }
}

// ---------------------------------------------------------------------------
// 4) Generic WMMA fp32 GEMM: C[M,N] = A[M,Kd] @ W[Kd,N] (+bias).
//    Kd % 4 == 0, N % 16 == 0, M % 16 == 0. One 16x16 tile per wave.
// ---------------------------------------------------------------------------
__global__ __launch_bounds__(256) void gemm_wmma_kernel(
    const float* __restrict__ A, const float* __restrict__ W,
    const float* __restrict__ bias, float* __restrict__ C,
    int M, int Kd, int Nn) {
  const int wave = threadIdx.x >> 5;
  const int lane = threadIdx.x & 31;
  const int tilesN = Nn >> 4;
  const int totalTiles = (M >> 4) * tilesN;
  const int tile = blockIdx.x * 8 + wave;
  if (tile >= totalTiles) return;                 // wave-uniform
  const int m0 = (tile / tilesN) << 4;
  const int n0 = (tile % tilesN) << 4;
  const int half = lane >> 4;                     // K-half select
  const int mr   = lane & 15;

  v8f acc = {};
  for (int k0 = 0; k0 < Kd; k0 += 4) {
    const int ka = k0 + half * 2;
    // A frag: row = m0+mr, k = ka, ka+1 (8B aligned since Kd%4==0, ka even)
    const float2 a2 = *(const float2*)&A[(size_t)(m0 + mr) * Kd + ka];
    v2f a; a[0] = a2.x; a[1] = a2.y;
    // B frag: W[k][n], n = n0+mr, k = ka, ka+1
    v2f bv; bv[0] = W[(size_t)ka * Nn + n0 + mr];
            bv[1] = W[(size_t)(ka + 1) * Nn + n0 + mr];
    acc = wmma_f32_16x16x4(a, bv, acc);
  }
  const float bb = bias ? bias[n0 + mr] : 0.f;
#pragma unroll
  for (int j = 0; j < 8; ++j) {
    const int row = m0 + j + half * 8;            // C/D layout: VGPR j -> M=j / j+8
    C[(size_t)row * Nn + n0 + mr] = acc[j] + bb;
  }
}

// ---------------------------------------------------------------------------
// 5) EdgeConv core: for 16 centers per WG,
//      h_out[c] = max_t( concat(rel[c,t], h_act[nbr[c,t]]) @ Wxn ) + bxn + xi[c]
//    WT (transposed, zero-padded) staged once in dynamic LDS (up to 265 KB).
//    A-tiles (16 rows x Kd4) are DOUBLE-BUFFERED: gathers for slot t+1 are
//    issued before the WMMA K-loop of slot t and committed to the alternate
//    buffer afterwards, so global-gather latency hides behind WMMA issue.
//    Max over the 20 slots = elementwise fmax of D frags (same lane mapping).
//    Total LDS <= 265KB + 2*16.6KB = 298.5KB -> fits CDNA5 320KB WGP LDS.
// ---------------------------------------------------------------------------
__global__ __launch_bounds__(256) void edge_conv_kernel(
    const float* __restrict__ h_in,   // [BN, ind] activated features
    const float* __restrict__ pos,    // [BN, 3]
    const int*   __restrict__ nbr,    // [BN, KNN] global neighbor ids
    const float* __restrict__ Wxn,    // [ind+3, out]
    const float* __restrict__ bxn,    // [out]
    const float* __restrict__ xi,     // [BN, out]  (h @ Wxi + bxi)
    float* __restrict__ h_out,        // [BN, out]
    int ind, int out) {
  extern __shared__ float lds[];
  const int Kd  = ind + 3;
  const int Kd4 = (Kd + 3) & ~3;
  float* WT  = lds;                         // [out][Kd4] transposed weights
  float* At0 = WT + (size_t)out * Kd4;      // [16][Kd4] A-tile buffer 0
  float* At1 = At0 + (size_t)16 * Kd4;      // [16][Kd4] A-tile buffer 1

  const int tid  = threadIdx.x;
  const int c0   = blockIdx.x * 16;         // center base
  const int wave = tid >> 5;
  const int lane = tid & 31;
  const int half = lane >> 4;
  const int mr   = lane & 15;
  const int strips = out >> 4;              // 16-wide N strips (4/8/16)

  // Fixed staging coordinates: each thread owns row srow, columns scol+16e.
  const int srow = tid >> 4;                // 0..15
  const int scol = tid & 15;

  // Stage WT once: WT[n][k] = Wxn[k][n], zero-pad k in [Kd, Kd4)
  for (int idx = tid; idx < out * Kd4; idx += 256) {
    const int n = idx / Kd4, k = idx % Kd4;
    WT[idx] = (k < Kd) ? Wxn[(size_t)k * out + n] : 0.f;
  }

  // Stage A-tile for slot 0 directly.
  {
    const int c = c0 + srow;
    const int g = nbr[c * KNN + 0];
#pragma unroll
    for (int e = 0; e < MAXE; ++e) {
      const int k = scol + e * 16;
      if (k < Kd4) {
        float v = 0.f;
        if (k < 3)       v = pos[g * 3 + k] - pos[c * 3 + k];
        else if (k < Kd) v = h_in[(size_t)g * ind + (k - 3)];
        At0[srow * Kd4 + k] = v;
      }
    }
  }

  v8f runmax0, runmax1;
#pragma unroll
  for (int j = 0; j < 8; ++j) { runmax0[j] = -3.0e38f; runmax1[j] = -3.0e38f; }

  __syncthreads();                          // WT + At0 ready

  float pv[MAXE];
  for (int t = 0; t < KNN; ++t) {
    const float* cur = (t & 1) ? At1 : At0;
    float*       nxt = (t & 1) ? At0 : At1;
    const bool more = (t + 1) < KNN;

    // ---- issue gathers for slot t+1 (loads only; no waits needed yet) ----
    if (more) {
      const int c = c0 + srow;
      const int g = nbr[c * KNN + (t + 1)];
#pragma unroll
      for (int e = 0; e < MAXE; ++e) {
        const int k = scol + e * 16;
        float v = 0.f;
        if (k < Kd4) {
          if (k < 3)       v = pos[g * 3 + k] - pos[c * 3 + k];
          else if (k < Kd) v = h_in[(size_t)g * ind + (k - 3)];
        }
        pv[e] = v;
      }
    }

    // ---- WMMA K-loop on current buffer (hides the gather latency) -------
#pragma unroll
    for (int si = 0; si < 2; ++si) {
      const int s = wave + si * 8;
      if (s >= strips) break;               // wave-uniform
      const int n0 = s << 4;
      v8f acc = {};
      for (int k0 = 0; k0 < Kd4; k0 += 4) {
        const int ka = k0 + half * 2;
        const float2 a2 = *(const float2*)&cur[mr * Kd4 + ka];         // ds b64
        const float2 b2 = *(const float2*)&WT[(n0 + mr) * Kd4 + ka];   // ds b64
        v2f a; a[0] = a2.x; a[1] = a2.y;
        v2f b; b[0] = b2.x; b[1] = b2.y;
        acc = wmma_f32_16x16x4(a, b, acc);
      }
      if (si == 0) {
#pragma unroll
        for (int j = 0; j < 8; ++j) runmax0[j] = fmaxf(runmax0[j], acc[j]);
      } else {
#pragma unroll
        for (int j = 0; j < 8; ++j) runmax1[j] = fmaxf(runmax1[j], acc[j]);
      }
    }

    // ---- commit prefetched slot t+1 to the alternate buffer --------------
    if (more) {
#pragma unroll
      for (int e = 0; e < MAXE; ++e) {
        const int k = scol + e * 16;
        if (k < Kd4) nxt[srow * Kd4 + k] = pv[e];
      }
    }
    __syncthreads();                        // nxt complete; cur reads done
  }

  // Epilogue: h_out = runmax + bxn + xi
#pragma unroll
  for (int si = 0; si < 2; ++si) {
    const int s = wave + si * 8;
    if (s >= strips) break;
    const int n = (s << 4) + mr;
    const float bb = bxn[n];
#pragma unroll
    for (int j = 0; j < 8; ++j) {
      const int c = c0 + j + half * 8;
      const float rv = (si == 0) ? runmax0[j] : runmax1[j];
      h_out[(size_t)c * out + n] = rv + bb + xi[(size_t)c * out + n];
    }
  }
}

// ---------------------------------------------------------------------------
// 6) Final 128 -> 5 linear with bias (N too small for WMMA).
// ---------------------------------------------------------------------------
__global__ __launch_bounds__(256) void out_linear_kernel(
    const float* __restrict__ h, const float* __restrict__ Wo,
    const float* __restrict__ bo, float* __restrict__ y) {
  const int r = blockIdx.x * 256 + threadIdx.x;
  if (r >= BNPTS) return;
  float acc[5];
#pragma unroll
  for (int j = 0; j < 5; ++j) acc[j] = bo[j];
  for (int k = 0; k < 128; ++k) {
    const float v = h[(size_t)r * 128 + k];
#pragma unroll
    for (int j = 0; j < 5; ++j) acc[j] = fmaf(v, Wo[k * 5 + j], acc[j]);
  }
#pragma unroll
  for (int j = 0; j < 5; ++j) y[(size_t)r * 5 + j] = acc[j];
}

// ---------------------------------------------------------------------------
// Orchestration
// ---------------------------------------------------------------------------
static inline size_t edge_lds_bytes(int ind, int out) {
  const int Kd4 = ((ind + 3) + 3) & ~3;
  return (size_t)(out + 32) * Kd4 * sizeof(float);   // WT + 2 A-tile buffers
}

extern "C" void kernel_launch(void* const* d_in, const int* in_sizes, int n_in,
                              void* d_out, int out_size, void* d_ws, size_t ws_size,
                              hipStream_t stream) {
  (void)in_sizes; (void)n_in; (void)out_size; (void)ws_size;
  const float* pos = (const float*)d_in[0];
  const float* x   = (const float*)d_in[1];
  // params in setup_inputs() dict order:
  const float* Wxi0 = (const float*)d_in[2];  const float* bxi0 = (const float*)d_in[3];
  const float* Wxn0 = (const float*)d_in[4];  const float* bxn0 = (const float*)d_in[5];
  const float* bn1g = (const float*)d_in[6];  const float* bn1b = (const float*)d_in[7];
  const float* Wxi1 = (const float*)d_in[8];  const float* bxi1 = (const float*)d_in[9];
  const float* Wxn1 = (const float*)d_in[10]; const float* bxn1 = (const float*)d_in[11];
  const float* bn2g = (const float*)d_in[12]; const float* bn2b = (const float*)d_in[13];
  const float* Wxi2 = (const float*)d_in[14]; const float* bxi2 = (const float*)d_in[15];
  const float* Wxn2 = (const float*)d_in[16]; const float* bxn2 = (const float*)d_in[17];
  const float* bn3g = (const float*)d_in[18]; const float* bn3b = (const float*)d_in[19];
  const float* Wxi3 = (const float*)d_in[20]; const float* bxi3 = (const float*)d_in[21];
  const float* Wxn3 = (const float*)d_in[22]; const float* bxn3 = (const float*)d_in[23];
  const float* Wseg0 = (const float*)d_in[24];
  const float* bns0g = (const float*)d_in[25]; const float* bns0b = (const float*)d_in[26];
  const float* Wseg1 = (const float*)d_in[27];
  const float* bns1g = (const float*)d_in[28]; const float* bns1b = (const float*)d_in[29];
  const float* Wout = (const float*)d_in[30]; const float* bout = (const float*)d_in[31];

  // Workspace layout
  int*   nbr   = (int*)d_ws;                                  // BN*K ints
  float* h_a   = (float*)((char*)d_ws + (size_t)BNPTS * KNN * sizeof(int));
  float* h_b   = h_a + (size_t)BNPTS * 256;
  float* xi    = h_b + (size_t)BNPTS * 256;
  float* scale = xi  + (size_t)BNPTS * 256;                   // 256
  float* shift = scale + 256;                                 // 256

  const int ec_grid = BNPTS / 16;                             // 2048

  // ---- kNN graph -----------------------------------------------------------
  knn_kernel<<<NUM_B * (NUM_N / 256), 256, 0, stream>>>(pos, nbr);

  // ---- layer 0: 4 -> 64 (no pre-act) --------------------------------------
  {
    const int ind = 4, out = 64;
    const int tiles = (BNPTS / 16) * (out / 16);
    gemm_wmma_kernel<<<tiles / 8, 256, 0, stream>>>(x, Wxi0, bxi0, xi, BNPTS, ind, out);
    edge_conv_kernel<<<ec_grid, 256, edge_lds_bytes(ind, out), stream>>>(
        x, pos, nbr, Wxn0, bxn0, xi, h_a, ind, out);
  }
  // ---- layer 1: 64 -> 128 --------------------------------------------------
  {
    const int ind = 64, out = 128;
    bn_stats_kernel<<<ind, 256, 0, stream>>>(h_a, bn1g, bn1b, scale, shift, ind);
    bn_relu_kernel<<<(BNPTS * ind + 255) / 256, 256, 0, stream>>>(h_a, scale, shift, h_b, ind);
    const int tiles = (BNPTS / 16) * (out / 16);
    gemm_wmma_kernel<<<tiles / 8, 256, 0, stream>>>(h_b, Wxi1, bxi1, xi, BNPTS, ind, out);
    edge_conv_kernel<<<ec_grid, 256, edge_lds_bytes(ind, out), stream>>>(
        h_b, pos, nbr, Wxn1, bxn1, xi, h_a, ind, out);
  }
  // ---- layer 2: 128 -> 256 -------------------------------------------------
  {
    const int ind = 128, out = 256;
    bn_stats_kernel<<<ind, 256, 0, stream>>>(h_a, bn2g, bn2b, scale, shift, ind);
    bn_relu_kernel<<<(BNPTS * ind + 255) / 256, 256, 0, stream>>>(h_a, scale, shift, h_b, ind);
    const int tiles = (BNPTS / 16) * (out / 16);
    gemm_wmma_kernel<<<tiles / 8, 256, 0, stream>>>(h_b, Wxi2, bxi2, xi, BNPTS, ind, out);
    edge_conv_kernel<<<ec_grid, 256, edge_lds_bytes(ind, out), stream>>>(
        h_b, pos, nbr, Wxn2, bxn2, xi, h_a, ind, out);
  }
  // ---- layer 3: 256 -> 256 -------------------------------------------------
  {
    const int ind = 256, out = 256;
    bn_stats_kernel<<<ind, 256, 0, stream>>>(h_a, bn3g, bn3b, scale, shift, ind);
    bn_relu_kernel<<<(BNPTS * ind + 255) / 256, 256, 0, stream>>>(h_a, scale, shift, h_b, ind);
    const int tiles = (BNPTS / 16) * (out / 16);
    gemm_wmma_kernel<<<tiles / 8, 256, 0, stream>>>(h_b, Wxi3, bxi3, xi, BNPTS, ind, out);
    edge_conv_kernel<<<ec_grid, 256, edge_lds_bytes(ind, out), stream>>>(
        h_b, pos, nbr, Wxn3, bxn3, xi, h_a, ind, out);
  }
  // ---- seg head: 256 -> 256 (BN+ReLU) -------------------------------------
  {
    const int tiles = (BNPTS / 16) * (256 / 16);
    gemm_wmma_kernel<<<tiles / 8, 256, 0, stream>>>(h_a, Wseg0, nullptr, h_b, BNPTS, 256, 256);
    bn_stats_kernel<<<256, 256, 0, stream>>>(h_b, bns0g, bns0b, scale, shift, 256);
    bn_relu_kernel<<<(BNPTS * 256 + 255) / 256, 256, 0, stream>>>(h_b, scale, shift, h_a, 256);
  }
  // ---- seg head: 256 -> 128 (BN+ReLU) -------------------------------------
  {
    const int tiles = (BNPTS / 16) * (128 / 16);
    gemm_wmma_kernel<<<tiles / 8, 256, 0, stream>>>(h_a, Wseg1, nullptr, h_b, BNPTS, 256, 128);
    bn_stats_kernel<<<128, 256, 0, stream>>>(h_b, bns1g, bns1b, scale, shift, 128);
    bn_relu_kernel<<<(BNPTS * 128 + 255) / 256, 256, 0, stream>>>(h_b, scale, shift, h_a, 128);
  }
  // ---- output: 128 -> 5 ----------------------------------------------------
  out_linear_kernel<<<BNPTS / 256, 256, 0, stream>>>(h_a, Wout, bout, (float*)d_out);
}